// DeepRMPNetwork_90280212562365
// MI455X (gfx1250) — compile-verified
//
#include <hip/hip_runtime.h>
#include <math.h>

#define DIM     24
#define HID     256
#define NUM_LO  276
#define NSAMP   4096
#define NEGS    (-0.01f)
#define NPAD    32               // padded N (columns of J1/J2)
#define MROWS2  320              // 276 (WLo) + 24 (WLd) padded to 20*16 -> 40 tiles, 5/wave
#define KC      32               // K-chunk staged in LDS

typedef float v2f __attribute__((ext_vector_type(2)));
typedef float v4f __attribute__((ext_vector_type(4)));
typedef float v8f __attribute__((ext_vector_type(8)));

struct Shm {
  alignas(16) float qs[DIM];
  alignas(16) float qds[DIM];
  alignas(16) float h1s[HID];
  alignas(16) float dR1s[HID];
  alignas(16) float h2s[HID];
  alignas(16) float dR2s[HID];
  alignas(16) float sig_s[DIM];
  alignas(16) float Ld_s[DIM];
  alignas(16) float L_s[DIM*DIM];
  int   trs[NUM_LO], tcs[NUM_LO];
  alignas(16) float J1T[NPAD*HID];     // J1 transposed: [n][k], rows n>=24 zero
  alignas(16) float J2T[NPAD*HID];     // J2 transposed: [n][m], rows n>=24 zero
  alignas(16) float pan[MROWS2*KC];    // A-panel staging (reused for qdm/um at the end)
  alignas(16) float dlos[NUM_LO*DIM];  // [276][24] row-major flat (faithful flat reindex)
  alignas(16) float dlds[DIM*DIM];     // dld_dq [24][24]
  alignas(16) float dldts[DIM];
  alignas(16) float dlots[NUM_LO];
  alignas(16) float dLts[DIM*DIM];
  alignas(16) float dMts[DIM*DIM];
};

// 256-long dot product: float4 global reads + v4f LDS reads
__device__ __forceinline__ float dot256(const float* __restrict__ w, const float* hs) {
  float acc = 0.f;
  const float4* wv = (const float4*)w;
  #pragma unroll 8
  for (int j = 0; j < HID/4; ++j) {
    float4 a = wv[j];
    v4f h = *(const v4f*)(hs + 4*j);
    acc += a.x*h[0] + a.y*h[1] + a.z*h[2] + a.w*h[3];
  }
  return acc;
}

__global__ __launch_bounds__(256)
void rmp_kernel(const float* __restrict__ q,   const float* __restrict__ qdot,
                const float* __restrict__ W1,  const float* __restrict__ b1,
                const float* __restrict__ W2,  const float* __restrict__ b2,
                const float* __restrict__ WG,  const float* __restrict__ bG,
                const float* __restrict__ WLd, const float* __restrict__ bLd,
                const float* __restrict__ WLo, const float* __restrict__ bLo,
                float* __restrict__ outM, float* __restrict__ outC,
                float* __restrict__ outG, float* __restrict__ u_ws, int phase)
{
  __shared__ Shm s;
  const int i    = blockIdx.x;
  const int tid  = threadIdx.x;
  const int lane = tid & 31;
  const int wave = tid >> 5;

  // strict-lower (row-major) index tables
  if (tid == 0) {
    int k = 0;
    for (int a = 1; a < DIM; ++a)
      for (int b = 0; b < a; ++b) { s.trs[k] = a; s.tcs[k] = b; ++k; }
  }
  if (tid < DIM) { s.qs[tid] = q[i*DIM + tid]; s.qds[tid] = qdot[i*DIM + tid]; }
  for (int o = tid; o < DIM*DIM; o += 256) s.L_s[o] = 0.0f;
  __syncthreads();

  // ---- trunk: h1 = lrelu(W1 q + b1) ----
  {
    float acc = b1[tid];
    #pragma unroll
    for (int d = 0; d < DIM; ++d) acc += W1[tid*DIM + d] * s.qs[d];
    s.h1s[tid]  = (acc > 0.f) ? acc : 0.01f * acc;
    s.dR1s[tid] = (acc > 0.f) ? 1.0f : NEGS;
  }
  __syncthreads();
  // ---- h2 = lrelu(W2 h1 + b2) ----
  {
    float acc = b2[tid] + dot256(W2 + (size_t)tid*HID, s.h1s);
    s.h2s[tid]  = (acc > 0.f) ? acc : 0.01f * acc;
    s.dR2s[tid] = (acc > 0.f) ? 1.0f : NEGS;
  }
  __syncthreads();
  // ---- heads: h3/Ld/sig (24), g (24), Lo -> strict-lower L (276) ----
  for (int o = tid; o < 2*DIM + NUM_LO; o += 256) {
    if (o < DIM) {
      float acc = bLd[o] + dot256(WLd + (size_t)o*HID, s.h2s);
      s.sig_s[o] = 1.0f / (1.0f + expf(-acc));
      s.Ld_s[o]  = (acc > 20.f) ? acc : log1pf(expf(acc));   // softplus
    } else if (o < 2*DIM) {
      int d0 = o - DIM;
      float acc = bG[d0] + dot256(WG + (size_t)d0*HID, s.h2s);
      if (phase == 0) outG[i*DIM + d0] = acc;
    } else {
      int k = o - 2*DIM;
      float acc = bLo[k] + dot256(WLo + (size_t)k*HID, s.h2s);
      s.L_s[s.trs[k]*DIM + s.tcs[k]] = acc;
    }
  }
  __syncthreads();
  if (tid < DIM) s.L_s[tid*DIM + tid] = s.Ld_s[tid];
  __syncthreads();

  if (phase == 0) {
    // M = L L^T + 1e-9 I ; u = L^T q_dot
    for (int o = tid; o < DIM*DIM; o += 256) {
      int a = o / DIM, b = o % DIM;
      float acc = (a == b) ? 1e-9f : 0.0f;
      for (int c0 = 0; c0 < DIM; ++c0) acc += s.L_s[a*DIM + c0] * s.L_s[b*DIM + c0];
      outM[(size_t)i*DIM*DIM + o] = acc;
    }
    if (tid < DIM) {
      float acc = 0.f;
      for (int a = 0; a < DIM; ++a) acc += s.L_s[a*DIM + tid] * s.qds[a];
      u_ws[i*DIM + tid] = acc;
    }
    return;
  }

  // =================== phase 1: Jacobian chain ===================
  // J1^T: [n][k], n-padded with zeros (so WMMA B-frag loads need no guards)
  for (int o = tid; o < NPAD*HID; o += 256) {
    int n = o >> 8, k = o & (HID-1);
    s.J1T[o] = (n < DIM) ? s.dR1s[k] * W1[k*DIM + n] : 0.0f;
  }
  __syncthreads();

  const int mrow  = lane & 15;
  const int ncol  = lane & 15;
  const int koff  = (lane >> 4) * 2;
  const int mhalf = (lane >> 4) * 8;

  // ---- GEMM1: T2 = W2 @ J1 ; J2 = dR2 (.) T2  (f32 WMMA, 16x16x4) ----
  {
    v8f acc[4];
    #pragma unroll
    for (int t = 0; t < 4; ++t) acc[t] = v8f{0.f,0.f,0.f,0.f,0.f,0.f,0.f,0.f};
    for (int kc = 0; kc < HID; kc += KC) {
      for (int o = tid; o < HID*KC; o += 256) {
        int m = o >> 5, kk = o & 31;
        s.pan[o] = W2[m*HID + kc + kk];
      }
      if (kc + KC < HID) __builtin_prefetch(&W2[(size_t)tid * HID + kc + KC], 0, 0);
      __syncthreads();
      for (int ks = 0; ks < KC; ks += 4) {
        #pragma unroll
        for (int t = 0; t < 4; ++t) {
          int tile = wave + t*8;                   // 32 tiles: 16 M-tiles x 2 N-tiles
          int mt = tile >> 1, nt = tile & 1;
          int n  = nt*16 + ncol;
          v2f a = *(const v2f*)&s.pan[(mt*16 + mrow)*KC + ks + koff];
          v2f b = *(const v2f*)&s.J1T[n*HID + kc + ks + koff];
          acc[t] = __builtin_amdgcn_wmma_f32_16x16x4_f32(false, a, false, b,
                                                         (short)0, acc[t], false, false);
        }
      }
      __syncthreads();
    }
    // write J2^T (pad rows n>=24 get exact zeros: their accumulators are zero)
    #pragma unroll
    for (int t = 0; t < 4; ++t) {
      int tile = wave + t*8;
      int mt = tile >> 1, nt = tile & 1;
      int n  = nt*16 + ncol;
      int m0 = mt*16 + mhalf;
      v4f lo = { s.dR2s[m0+0]*acc[t][0], s.dR2s[m0+1]*acc[t][1],
                 s.dR2s[m0+2]*acc[t][2], s.dR2s[m0+3]*acc[t][3] };
      v4f hi = { s.dR2s[m0+4]*acc[t][4], s.dR2s[m0+5]*acc[t][5],
                 s.dR2s[m0+6]*acc[t][6], s.dR2s[m0+7]*acc[t][7] };
      *(v4f*)&s.J2T[n*HID + m0]     = lo;
      *(v4f*)&s.J2T[n*HID + m0 + 4] = hi;
    }
  }
  __syncthreads();

  // ---- GEMM2/3 fused: [WLo;WLd;0] @ J2 -> dlo_dq (276x24), dld_dq = sig (.) (24x24) ----
  {
    v8f acc[5];
    #pragma unroll
    for (int t = 0; t < 5; ++t) acc[t] = v8f{0.f,0.f,0.f,0.f,0.f,0.f,0.f,0.f};
    for (int kc = 0; kc < HID; kc += KC) {
      for (int o = tid; o < MROWS2*KC; o += 256) {
        int m = o >> 5, kk = o & 31;
        int kg = kc + kk;
        float v;
        if (m < NUM_LO)            v = WLo[m*HID + kg];
        else if (m < NUM_LO + DIM) v = WLd[(m - NUM_LO)*HID + kg];
        else                       v = 0.0f;
        s.pan[o] = v;
      }
      __syncthreads();
      for (int ks = 0; ks < KC; ks += 4) {
        #pragma unroll
        for (int t = 0; t < 5; ++t) {
          int tile = wave + t*8;                  // 40 tiles: 20 M-tiles x 2 N-tiles
          int mt = tile >> 1, nt = tile & 1;
          int n  = nt*16 + ncol;
          v2f a = *(const v2f*)&s.pan[(mt*16 + mrow)*KC + ks + koff];
          v2f b = *(const v2f*)&s.J2T[n*HID + kc + ks + koff];
          acc[t] = __builtin_amdgcn_wmma_f32_16x16x4_f32(false, a, false, b,
                                                         (short)0, acc[t], false, false);
        }
      }
      __syncthreads();
    }
    #pragma unroll
    for (int t = 0; t < 5; ++t) {
      int tile = wave + t*8;
      int mt = tile >> 1, nt = tile & 1;
      int n  = nt*16 + ncol;
      if (n < DIM) {
        #pragma unroll
        for (int v = 0; v < 8; ++v) {
          int m = mt*16 + mhalf + v;
          if (m < NUM_LO) s.dlos[m*DIM + n] = acc[t][v];
          else if (m < NUM_LO + DIM) {
            int r = m - NUM_LO;
            s.dlds[r*DIM + n] = s.sig_s[r] * acc[t][v];
          }
        }
      }
    }
  }
  __syncthreads();

  // ---- dld_dt, dlo_dt ; assemble dL_dt ----
  for (int o = tid; o < DIM + NUM_LO; o += 256) {
    if (o < DIM) {
      float acc = 0.f;
      for (int n = 0; n < DIM; ++n) acc += s.dlds[o*DIM + n] * s.qds[n];
      s.dldts[o] = acc;
    } else {
      int k = o - DIM;
      float acc = 0.f;
      for (int n = 0; n < DIM; ++n) acc += s.dlos[k*DIM + n] * s.qds[n];
      s.dlots[k] = acc;
    }
  }
  for (int o = tid; o < DIM*DIM; o += 256) s.dLts[o] = 0.f;
  __syncthreads();
  for (int o = tid; o < NUM_LO; o += 256) s.dLts[s.trs[o]*DIM + s.tcs[o]] = s.dlots[o];
  if (tid < DIM) s.dLts[tid*DIM + tid] = s.dldts[tid];
  __syncthreads();

  // ---- dM_dt = L dL_dt^T + dL_dt L^T ----
  for (int o = tid; o < DIM*DIM; o += 256) {
    int a = o / DIM, b = o % DIM;
    float acc = 0.f;
    for (int c0 = 0; c0 < DIM; ++c0)
      acc += s.L_s[a*DIM + c0] * s.dLts[b*DIM + c0]
           + s.dLts[a*DIM + c0] * s.L_s[b*DIM + c0];
    s.dMts[o] = acc;
  }
  __syncthreads();

  // ---- c[i][j] = dM_dt@qd  -  sum_{a,b} dL_dq[i,j][a,b] * qd[m][a] * u[m][b] ----
  // (0.5*quad with quad = 2*u.v folded in; m = (24*i + j) % 4096, torch-faithful)
  if (tid < DIM) {
    int j = tid;
    int bidx = i*DIM + j;
    int m = bidx % NSAMP;
    float* qdm = &s.pan[j*64];        // reuse pan as per-thread scratch (no races)
    float* um  = &s.pan[j*64 + 32];
    for (int a = 0; a < DIM; ++a) {
      qdm[a] = qdot[m*DIM + a];
      um[a]  = u_ws[m*DIM + a];
    }
    float term1 = 0.f;
    for (int b = 0; b < DIM; ++b) term1 += s.dMts[j*DIM + b] * s.qds[b];
    float quad = 0.f;
    // strict-lower part: flat dlo buffer re-read at offset j*276+k (faithful reshape bug)
    for (int k = 0; k < NUM_LO; ++k)
      quad += s.dlos[j*NUM_LO + k] * qdm[s.trs[k]] * um[s.tcs[k]];
    // diagonal part: dL_dq[i,j][a,a] = dld_dq[i][a][j]
    for (int a = 0; a < DIM; ++a)
      quad += s.dlds[a*DIM + j] * qdm[a] * um[a];
    outC[bidx] = term1 - quad;
  }
}

extern "C" void kernel_launch(void* const* d_in, const int* in_sizes, int n_in,
                              void* d_out, int out_size, void* d_ws, size_t ws_size,
                              hipStream_t stream) {
  const float* q    = (const float*)d_in[0];
  const float* qdot = (const float*)d_in[1];
  const float* W1   = (const float*)d_in[2];
  const float* b1   = (const float*)d_in[3];
  const float* W2   = (const float*)d_in[4];
  const float* b2   = (const float*)d_in[5];
  const float* WG   = (const float*)d_in[6];
  const float* bG   = (const float*)d_in[7];
  const float* WLd  = (const float*)d_in[8];
  const float* bLd  = (const float*)d_in[9];
  const float* WLo  = (const float*)d_in[10];
  const float* bLo  = (const float*)d_in[11];

  float* outM = (float*)d_out;                               // [4096,24,24]
  float* outC = outM + (size_t)NSAMP*DIM*DIM;                // [4096,24]
  float* outG = outC + (size_t)NSAMP*DIM;                    // [4096,24]
  float* u_ws = (float*)d_ws;                                // [4096,24] = 393KB

  dim3 grid(NSAMP), block(256);
  rmp_kernel<<<grid, block, 0, stream>>>(q, qdot, W1, b1, W2, b2, WG, bG,
                                         WLd, bLd, WLo, bLo,
                                         outM, outC, outG, u_ws, /*phase=*/0);
  rmp_kernel<<<grid, block, 0, stream>>>(q, qdot, W1, b1, W2, b2, WG, bG,
                                         WLd, bLd, WLo, bLo,
                                         outM, outC, outG, u_ws, /*phase=*/1);
}